// GIN_53145925321200
// MI455X (gfx1250) — compile-verified
//
#include <hip/hip_runtime.h>
#include <hip/hip_bf16.h>

typedef float v2f __attribute__((ext_vector_type(2)));
typedef float v8f __attribute__((ext_vector_type(8)));

#define HID 96

// ---------------------------------------------------------------------------
// hpre = (1 + eps) * x        (eps is a device scalar), float4 vectorized
// ---------------------------------------------------------------------------
__global__ void scale_init(const float* __restrict__ x,
                           const float* __restrict__ eps,
                           float* __restrict__ y, int n4) {
  int i = blockIdx.x * blockDim.x + threadIdx.x;
  if (i >= n4) return;
  float s = 1.0f + eps[0];
  const float4* xv = (const float4*)x;
  float4* yv = (float4*)y;
  float4 v = xv[i];
  v.x *= s; v.y *= s; v.z *= s; v.w *= s;
  yv[i] = v;
}

// ---------------------------------------------------------------------------
// agg[dst[e]*96 + c] += feat[src[e]*96 + c]
// blockDim = (96, EDGES_PER_BLOCK): 96 = 3 full wave32s per edge, coalesced.
// ---------------------------------------------------------------------------
#define EDGES_PER_BLOCK 4
__global__ void edge_aggregate(const float* __restrict__ feat,
                               const int* __restrict__ src,
                               const int* __restrict__ dst,
                               float* __restrict__ agg, int E) {
  int e = blockIdx.x * EDGES_PER_BLOCK + threadIdx.y;
  if (e >= E) return;
  int s = src[e];
  int d = dst[e];
  int c = threadIdx.x;  // 0..95
  atomicAdd(&agg[d * HID + c], feat[s * HID + c]);
}

// ---------------------------------------------------------------------------
// out[M x 96] = relu(A[M x 96] @ W[96 x 96] + b[96])
// One block per 16-row M tile; 6 waves, wave w owns N-tile [16w, 16w+16).
// fp32 WMMA 16x16x4 (exact fp32 math; workload is memory bound so no reason
// to downcast). Fragment layouts per CDNA5 ISA 7.12.2.
// ---------------------------------------------------------------------------
__global__ void gemm_bias_relu(const float* __restrict__ A,
                               const float* __restrict__ W,
                               const float* __restrict__ b,
                               float* __restrict__ out, int M) {
  const int m0   = blockIdx.x * 16;
  const int wave = threadIdx.x >> 5;   // 0..5
  const int lane = threadIdx.x & 31;
  const int half = lane >> 4;          // 0: lanes 0-15, 1: lanes 16-31
  const int l    = lane & 15;
  const int n0   = wave * 16;

  // Stage the 16 x 96 A tile once into LDS; all 6 waves reuse it.
  __shared__ float As[16 * HID];
  for (int i = threadIdx.x; i < 16 * HID; i += blockDim.x) {
    int r = i / HID, c = i % HID;
    int row = m0 + r;
    As[i] = (row < M) ? A[row * HID + c] : 0.0f;
  }
  __syncthreads();

  v8f acc = {};
#pragma unroll
  for (int k0 = 0; k0 < HID; k0 += 4) {
    // A frag 16x4 (MxK): lane half selects K pair {0,1} vs {2,3}, row = l
    v2f af;
    af.x = As[l * HID + k0 + 2 * half + 0];
    af.y = As[l * HID + k0 + 2 * half + 1];
    // B frag 4x16 (KxN): col = n0 + l, lane half selects K pair
    v2f bf;
    bf.x = W[(k0 + 2 * half + 0) * HID + n0 + l];
    bf.y = W[(k0 + 2 * half + 1) * HID + n0 + l];
    // 8 args: (neg_a, A, neg_b, B, c_mod, C, reuse_a, reuse_b)
    acc = __builtin_amdgcn_wmma_f32_16x16x4_f32(false, af, false, bf,
                                                (short)0, acc, false, false);
  }

  // Epilogue: D VGPR r -> row m0 + r + 8*half, col n0 + l. Fuse bias + ReLU.
  float bias = b[n0 + l];
#pragma unroll
  for (int r = 0; r < 8; ++r) {
    int row = m0 + r + 8 * half;
    if (row < M) {
      float v = acc[r] + bias;
      out[row * HID + n0 + l] = v > 0.0f ? v : 0.0f;
    }
  }
}

// ---------------------------------------------------------------------------
// Inputs (setup_inputs order):
//  0 feat [50000*96] f32     1 src [E] i32        2 dst [E] i32
//  3 eps1 [1] f32   4 W1a [96*96]  5 b1a [96]  6 W1b [96*96]  7 b1b [96]
//  8 eps2 [1] f32   9 W2a [96*96] 10 b2a [96] 11 W2b [96*96] 12 b2b [96]
// ---------------------------------------------------------------------------
extern "C" void kernel_launch(void* const* d_in, const int* in_sizes, int n_in,
                              void* d_out, int out_size, void* d_ws, size_t ws_size,
                              hipStream_t stream) {
  const float* feat = (const float*)d_in[0];
  const int*   src  = (const int*)d_in[1];
  const int*   dst  = (const int*)d_in[2];
  const float* eps1 = (const float*)d_in[3];
  const float* W1a  = (const float*)d_in[4];
  const float* b1a  = (const float*)d_in[5];
  const float* W1b  = (const float*)d_in[6];
  const float* b1b  = (const float*)d_in[7];
  const float* eps2 = (const float*)d_in[8];
  const float* W2a  = (const float*)d_in[9];
  const float* b2a  = (const float*)d_in[10];
  const float* W2b  = (const float*)d_in[11];
  const float* b2b  = (const float*)d_in[12];

  const int nfeat = in_sizes[0];        // 50000 * 96
  const int M     = nfeat / HID;        // 50000
  const int E     = in_sizes[1];        // 800000

  float* hpre = (float*)d_ws;           // 19.2 MB
  float* t0   = hpre + (size_t)nfeat;   // 19.2 MB
  float* h1   = t0 + (size_t)nfeat;     // 19.2 MB

  const int n4 = nfeat / 4;
  dim3 initGrid((n4 + 255) / 256), initBlk(256);
  dim3 edgeGrid((E + EDGES_PER_BLOCK - 1) / EDGES_PER_BLOCK), edgeBlk(HID, EDGES_PER_BLOCK);
  dim3 gemmGrid((M + 15) / 16), gemmBlk(192);   // 6 waves

  // ---- Layer 1 ----
  scale_init<<<initGrid, initBlk, 0, stream>>>(feat, eps1, hpre, n4);
  edge_aggregate<<<edgeGrid, edgeBlk, 0, stream>>>(feat, src, dst, hpre, E);
  gemm_bias_relu<<<gemmGrid, gemmBlk, 0, stream>>>(hpre, W1a, b1a, t0, M);
  gemm_bias_relu<<<gemmGrid, gemmBlk, 0, stream>>>(t0, W1b, b1b, h1, M);

  // ---- Layer 2 (reuse hpre / t0) ----
  scale_init<<<initGrid, initBlk, 0, stream>>>(h1, eps2, hpre, n4);
  edge_aggregate<<<edgeGrid, edgeBlk, 0, stream>>>(h1, src, dst, hpre, E);
  gemm_bias_relu<<<gemmGrid, gemmBlk, 0, stream>>>(hpre, W2a, b2a, t0, M);
  gemm_bias_relu<<<gemmGrid, gemmBlk, 0, stream>>>(t0, W2b, b2b, (float*)d_out, M);
}